// RWKV_LAYER_11175504904467
// MI455X (gfx1250) — compile-verified
//
#include <hip/hip_runtime.h>
#include <hip/hip_bf16.h>
#include <math.h>

// ---------------------------------------------------------------------------
// RWKV single-token forward, 24 layers, D=1024, H=4096, fp32.
// Bandwidth-bound (1.25 GB weights @ 23.3 TB/s => ~54us floor).
// Matvecs run through V_WMMA_F32_16X16X4_F32 fed from double-buffered
// LDS tiles; tile staging uses GLOBAL_LOAD_ASYNC_TO_LDS_B128 (ASYNCcnt)
// when the builtin exists, else batched global_load_b128 via registers.
// ---------------------------------------------------------------------------

#define NLAYER 24
#define DMODEL 1024
#define HFFN   4096

typedef __attribute__((ext_vector_type(2))) float v2f;
typedef __attribute__((ext_vector_type(8))) float v8f;
typedef __attribute__((ext_vector_type(4))) int   v4i;

#define MV_BLOCK_ROWS 128          // rows per workgroup (8 waves x 16 rows)
#define MV_KC         64           // K-chunk per tile
#define MV_LDS_STRIDE (MV_KC + 4)  // 68 floats: bank=(4*row+col)%64 conflict-free,
                                   // row stride 272B keeps 16B alignment
#define MV_MAX_K      4096

#if defined(__AMDGCN__) && __has_builtin(__builtin_amdgcn_global_load_async_to_lds_b128) && \
    __has_builtin(__builtin_amdgcn_s_wait_asynccnt)
#define RW_ASYNC_LDS 1
#else
#define RW_ASYNC_LDS 0
#endif

#if RW_ASYNC_LDS
__device__ __forceinline__ void rw_async_copy16(const float* g, float* l) {
    // Signature (from clang-22 diagnostics): param0 = int4 AS(1)*, param1 = int4 AS(3)*,
    // then two immediate ints (offset, cpol). Low 32 bits of a generic LDS
    // pointer are the LDS byte offset.
    auto* gp = (__attribute__((address_space(1))) v4i*)(uintptr_t)g;
    auto* lp = (__attribute__((address_space(3))) v4i*)(unsigned int)(uintptr_t)l;
    __builtin_amdgcn_global_load_async_to_lds_b128(gp, lp, 0, 0);
}
#endif

// act: 0 = identity, 1 = sigmoid, 2 = relu^2
// y[m] = (addv ? addv[m] : 0) + (mulv ? mulv[m] : 1) * act(dot(W[m,:], x))
__global__ __launch_bounds__(256) void rwkv_matvec_wmma(
    const float* __restrict__ W, const float* __restrict__ x,
    const float* __restrict__ addv, const float* __restrict__ mulv,
    float* __restrict__ y, int K, int act)
{
    __shared__ float As[2][MV_BLOCK_ROWS * MV_LDS_STRIDE];
    __shared__ float xs[MV_MAX_K];

    const int tid  = threadIdx.x;
    const int lane = tid & 31;
    const int wave = tid >> 5;
    const int hi   = lane >> 4;    // 0: lanes 0-15, 1: lanes 16-31
    const int lrow = lane & 15;
    const int rowBlock = blockIdx.x * MV_BLOCK_ROWS;

    // stage the whole x vector once (<= 16KB)
    for (int i = tid; i < K; i += 256) xs[i] = x[i];

    v8f acc = {0.f, 0.f, 0.f, 0.f, 0.f, 0.f, 0.f, 0.f};
    const int nchunk = K / MV_KC;

#if RW_ASYNC_LDS
    // ---- prologue: async-stage chunk 0 into buffer 0 ----
    {
        #pragma unroll
        for (int i = 0; i < 8; ++i) {
            int idx = tid + (i << 8);
            int r   = idx >> 4;
            int c4  = (idx & 15) << 2;
            rw_async_copy16(W + (size_t)(rowBlock + r) * (size_t)K + (size_t)c4,
                            &As[0][r * MV_LDS_STRIDE + c4]);
        }
    }
    for (int c = 0; c < nchunk; ++c) {
        __builtin_amdgcn_s_wait_asynccnt(0);   // my staged tile is in LDS
        __syncthreads();                        // everyone's tile is in LDS
        // stage next tile into the other buffer while computing this one
        if (c + 1 < nchunk) {
            const int k1 = (c + 1) * MV_KC;
            #pragma unroll
            for (int i = 0; i < 8; ++i) {
                int idx = tid + (i << 8);
                int r   = idx >> 4;
                int c4  = (idx & 15) << 2;
                rw_async_copy16(W + (size_t)(rowBlock + r) * (size_t)K + (size_t)(k1 + c4),
                                &As[(c + 1) & 1][r * MV_LDS_STRIDE + c4]);
            }
        }
        // ---- 16 WMMAs over this chunk ----
        // A frag (16x4 f32, ISA 7.12.2): lane holds row (lane&15);
        //   VGPR0 = K = kk+2*hi, VGPR1 = K = kk+2*hi+1. B replicated over N.
        const float* arow = &As[c & 1][(wave * 16 + lrow) * MV_LDS_STRIDE];
        const float* xrow = &xs[c * MV_KC];
        #pragma unroll
        for (int kk = 0; kk < MV_KC; kk += 4) {
            const float2 af = *reinterpret_cast<const float2*>(arow + kk + 2 * hi);
            v2f a; a.x = af.x; a.y = af.y;
            v2f b; b.x = xrow[kk + 2 * hi]; b.y = xrow[kk + 2 * hi + 1];
            acc = __builtin_amdgcn_wmma_f32_16x16x4_f32(
                false, a, false, b, (short)0, acc, false, false);
        }
    }
#else
    // ---- fallback: batched register staging, double-buffered ----
    float4 rg[8];
    {
        #pragma unroll
        for (int i = 0; i < 8; ++i) {
            int idx = tid + (i << 8);
            int r   = idx >> 4;
            int c4  = (idx & 15) << 2;
            rg[i] = *reinterpret_cast<const float4*>(
                W + (size_t)(rowBlock + r) * (size_t)K + (size_t)c4);
        }
        #pragma unroll
        for (int i = 0; i < 8; ++i) {
            int idx = tid + (i << 8);
            int r   = idx >> 4;
            int c4  = (idx & 15) << 2;
            *reinterpret_cast<float4*>(&As[0][r * MV_LDS_STRIDE + c4]) = rg[i];
        }
    }
    for (int c = 0; c < nchunk; ++c) {
        __syncthreads();
        // issue all 8 loads for the next tile up front (no per-load waits)
        if (c + 1 < nchunk) {
            const int k1 = (c + 1) * MV_KC;
            #pragma unroll
            for (int i = 0; i < 8; ++i) {
                int idx = tid + (i << 8);
                int r   = idx >> 4;
                int c4  = (idx & 15) << 2;
                rg[i] = *reinterpret_cast<const float4*>(
                    W + (size_t)(rowBlock + r) * (size_t)K + (size_t)(k1 + c4));
            }
        }
        const float* arow = &As[c & 1][(wave * 16 + lrow) * MV_LDS_STRIDE];
        const float* xrow = &xs[c * MV_KC];
        #pragma unroll
        for (int kk = 0; kk < MV_KC; kk += 4) {
            const float2 af = *reinterpret_cast<const float2*>(arow + kk + 2 * hi);
            v2f a; a.x = af.x; a.y = af.y;
            v2f b; b.x = xrow[kk + 2 * hi]; b.y = xrow[kk + 2 * hi + 1];
            acc = __builtin_amdgcn_wmma_f32_16x16x4_f32(
                false, a, false, b, (short)0, acc, false, false);
        }
        if (c + 1 < nchunk) {
            #pragma unroll
            for (int i = 0; i < 8; ++i) {
                int idx = tid + (i << 8);
                int r   = idx >> 4;
                int c4  = (idx & 15) << 2;
                *reinterpret_cast<float4*>(&As[(c + 1) & 1][r * MV_LDS_STRIDE + c4]) = rg[i];
            }
        }
    }
#endif

    // D layout (16x16 f32): VGPR j -> M=j (lanes 0-15) / M=j+8 (lanes 16-31),
    // N = lane&15. All N columns identical; lanes with N==0 write results.
    if (lrow == 0) {
        const int base = rowBlock + wave * 16 + hi * 8;
        #pragma unroll
        for (int j = 0; j < 8; ++j) {
            float d = acc[j];
            if (act == 1) {
                d = 1.f / (1.f + expf(-d));
            } else if (act == 2) {
                d = fmaxf(d, 0.f);
                d = d * d;
            }
            if (mulv) d *= mulv[base + j];
            if (addv) d += addv[base + j];
            y[base + j] = d;
        }
    }
}

// ---------------------------------------------------------------------------
// LayerNorm + token-shift mixing (SA branch: 3 mixed outputs).
// ---------------------------------------------------------------------------
__global__ __launch_bounds__(256) void rwkv_ln_mix3(
    const float* __restrict__ xc, const float* __restrict__ w, const float* __restrict__ b,
    const float* __restrict__ sprev, const float* __restrict__ tmk,
    const float* __restrict__ tmv, const float* __restrict__ tmr,
    float* __restrict__ xn_out, float* __restrict__ xk,
    float* __restrict__ xv, float* __restrict__ xr)
{
    __shared__ float sh1[256], sh2[256];
    const int tid = threadIdx.x;
    float s = 0.f, s2 = 0.f;
    #pragma unroll
    for (int i = tid; i < DMODEL; i += 256) {
        float v = xc[i];
        s += v; s2 += v * v;
    }
    sh1[tid] = s; sh2[tid] = s2;
    __syncthreads();
    for (int off = 128; off > 0; off >>= 1) {
        if (tid < off) { sh1[tid] += sh1[tid + off]; sh2[tid] += sh2[tid + off]; }
        __syncthreads();
    }
    const float mean = sh1[0] * (1.f / DMODEL);
    const float var  = sh2[0] * (1.f / DMODEL) - mean * mean;
    const float rstd = rsqrtf(var + 1e-5f);
    #pragma unroll
    for (int i = tid; i < DMODEL; i += 256) {
        float xn = (xc[i] - mean) * rstd * w[i] + b[i];
        float sp = sprev[i];
        xn_out[i] = xn;
        xk[i] = xn + sp * tmk[i];
        xv[i] = xn + sp * tmv[i];
        xr[i] = xn + sp * tmr[i];
    }
}

// FFN branch: 2 mixed outputs.
__global__ __launch_bounds__(256) void rwkv_ln_mix2(
    const float* __restrict__ xc, const float* __restrict__ w, const float* __restrict__ b,
    const float* __restrict__ sprev, const float* __restrict__ tmk,
    const float* __restrict__ tmr,
    float* __restrict__ xn_out, float* __restrict__ xk, float* __restrict__ xr)
{
    __shared__ float sh1[256], sh2[256];
    const int tid = threadIdx.x;
    float s = 0.f, s2 = 0.f;
    #pragma unroll
    for (int i = tid; i < DMODEL; i += 256) {
        float v = xc[i];
        s += v; s2 += v * v;
    }
    sh1[tid] = s; sh2[tid] = s2;
    __syncthreads();
    for (int off = 128; off > 0; off >>= 1) {
        if (tid < off) { sh1[tid] += sh1[tid + off]; sh2[tid] += sh2[tid + off]; }
        __syncthreads();
    }
    const float mean = sh1[0] * (1.f / DMODEL);
    const float var  = sh2[0] * (1.f / DMODEL) - mean * mean;
    const float rstd = rsqrtf(var + 1e-5f);
    #pragma unroll
    for (int i = tid; i < DMODEL; i += 256) {
        float xn = (xc[i] - mean) * rstd * w[i] + b[i];
        float sp = sprev[i];
        xn_out[i] = xn;
        xk[i] = xn + sp * tmk[i];
        xr[i] = xn + sp * tmr[i];
    }
}

// ---------------------------------------------------------------------------
// WKV numerically-stable state update (elementwise over D).
// ---------------------------------------------------------------------------
__global__ __launch_bounds__(256) void rwkv_wkv(
    const float* __restrict__ k, const float* __restrict__ v, const float* __restrict__ r,
    const float* __restrict__ tf, const float* __restrict__ td,
    const float* __restrict__ aa, const float* __restrict__ bb, const float* __restrict__ pp,
    float* __restrict__ naa, float* __restrict__ nbb, float* __restrict__ npp,
    float* __restrict__ t)
{
    const int i = blockIdx.x * blockDim.x + threadIdx.x;
    if (i >= DMODEL) return;
    const float ki = k[i], vi = v[i];
    const float ppi = pp[i], aai = aa[i], bbi = bb[i];

    const float ww = tf[i] + ki;
    const float pm = fmaxf(ppi, ww);
    const float e1 = expf(ppi - pm);
    const float e2 = expf(ww - pm);
    const float ab = (e1 * aai + e2 * vi) / (e1 * bbi + e2);
    const float rs = 1.f / (1.f + expf(-r[i]));
    t[i] = rs * ab;

    const float ww2 = ppi + td[i];
    const float p2  = fmaxf(ww2, ki);
    const float e1b = expf(ww2 - p2);
    const float e2b = expf(ki - p2);
    naa[i] = e1b * aai + e2b * vi;
    nbb[i] = e1b * bbi + e2b;
    npp[i] = p2;
}

// ---------------------------------------------------------------------------
// Host-side orchestration: serial per-layer chain on `stream`.
// ---------------------------------------------------------------------------
extern "C" void kernel_launch(void* const* d_in, const int* in_sizes, int n_in,
                              void* d_out, int out_size, void* d_ws, size_t ws_size,
                              hipStream_t stream) {
    (void)in_sizes; (void)n_in; (void)out_size; (void)ws_size;
    const int D = DMODEL, H = HFFN;

    const float* x_in   = (const float*)d_in[0];
    const float* state  = (const float*)d_in[1];
    const float* ln1w   = (const float*)d_in[2];
    const float* ln1b   = (const float*)d_in[3];
    const float* ln2w   = (const float*)d_in[4];
    const float* ln2b   = (const float*)d_in[5];
    const float* tdcy   = (const float*)d_in[6];
    const float* tfst   = (const float*)d_in[7];
    const float* tmk    = (const float*)d_in[8];
    const float* tmv    = (const float*)d_in[9];
    const float* tmr    = (const float*)d_in[10];
    const float* key_w  = (const float*)d_in[11];
    const float* val_w  = (const float*)d_in[12];
    const float* rec_w  = (const float*)d_in[13];
    const float* out_w  = (const float*)d_in[14];
    const float* tmkf   = (const float*)d_in[15];
    const float* tmrf   = (const float*)d_in[16];
    const float* keyf_w = (const float*)d_in[17];
    const float* recf_w = (const float*)d_in[18];
    const float* valf_w = (const float*)d_in[19];

    float* out       = (float*)d_out;       // [0:D) = xf
    float* state_out = out + D;             // [D : D + L*5*D) = new state

    float* ws  = (float*)d_ws;
    float* cur = ws;            // D
    float* xk  = ws + 1 * D;
    float* xv  = ws + 2 * D;
    float* xr  = ws + 3 * D;
    float* kb  = ws + 4 * D;
    float* vb  = ws + 5 * D;
    float* rb  = ws + 6 * D;
    float* tb  = ws + 7 * D;
    float* sx  = ws + 8 * D;
    float* xk2 = ws + 9 * D;
    float* xr2 = ws + 10 * D;
    float* r2  = ws + 11 * D;
    float* q   = ws + 12 * D;   // H floats

    (void)hipMemcpyAsync(cur, x_in, (size_t)D * sizeof(float),
                         hipMemcpyDeviceToDevice, stream);

    const dim3 blk(256);
    const dim3 gridD(D / MV_BLOCK_ROWS);   // 8 blocks  (M = 1024)
    const dim3 gridH(H / MV_BLOCK_ROWS);   // 32 blocks (M = 4096)

    for (int l = 0; l < NLAYER; ++l) {
        const float* st  = state     + (size_t)l * 5 * D;
        float*       sto = state_out + (size_t)l * 5 * D;
        const size_t wD = (size_t)l * D * D;
        const size_t wH = (size_t)l * H * D;

        // --- time-mix (SA) ---
        rwkv_ln_mix3<<<1, blk, 0, stream>>>(
            cur, ln1w + (size_t)l * D, ln1b + (size_t)l * D,
            st + D, tmk + (size_t)l * D, tmv + (size_t)l * D, tmr + (size_t)l * D,
            sto + D /* new xn state */, xk, xv, xr);

        rwkv_matvec_wmma<<<gridD, blk, 0, stream>>>(key_w + wD, xk, nullptr, nullptr, kb, D, 0);
        rwkv_matvec_wmma<<<gridD, blk, 0, stream>>>(val_w + wD, xv, nullptr, nullptr, vb, D, 0);
        rwkv_matvec_wmma<<<gridD, blk, 0, stream>>>(rec_w + wD, xr, nullptr, nullptr, rb, D, 0);

        rwkv_wkv<<<dim3(D / 256), blk, 0, stream>>>(
            kb, vb, rb, tfst + (size_t)l * D, tdcy + (size_t)l * D,
            st + 2 * D, st + 3 * D, st + 4 * D,
            sto + 2 * D, sto + 3 * D, sto + 4 * D, tb);

        // sx = xc + ow @ (sigmoid(r)*ab)
        rwkv_matvec_wmma<<<gridD, blk, 0, stream>>>(out_w + wD, tb, cur, nullptr, sx, D, 0);

        // --- channel-mix (FFN) ---
        rwkv_ln_mix2<<<1, blk, 0, stream>>>(
            sx, ln2w + (size_t)l * D, ln2b + (size_t)l * D,
            st /* old s0 */, tmkf + (size_t)l * D, tmrf + (size_t)l * D,
            sto /* new xn2 state */, xk2, xr2);

        // q = relu(kwf @ xk2)^2   (M=H, K=D)
        rwkv_matvec_wmma<<<gridH, blk, 0, stream>>>(keyf_w + wH, xk2, nullptr, nullptr, q, D, 2);
        // r2 = sigmoid(rwf @ xr2)
        rwkv_matvec_wmma<<<gridD, blk, 0, stream>>>(recf_w + wD, xr2, nullptr, nullptr, r2, D, 1);
        // out = sx + r2 * (vwf @ q)   (M=D, K=H)
        float* dst = (l == NLAYER - 1) ? out : cur;
        rwkv_matvec_wmma<<<gridD, blk, 0, stream>>>(valf_w + wH, q, sx, r2, dst, H, 0);
    }
}